// SDPLayer_19189913878957
// MI455X (gfx1250) — compile-verified
//
#include <hip/hip_runtime.h>
#include <hip/hip_bf16.h>

#define NN 1024
#define BM 128          // block tile rows
#define BN 64           // block tile cols
#define BK 32           // K tile
#define LDT 36          // LDS row stride (floats): even (b64-aligned), bank-spread

typedef __attribute__((ext_vector_type(2))) float v2f;
typedef __attribute__((ext_vector_type(8))) float v8f;

// CDNA5 async global->LDS path (no VGPR staging, tracked by ASYNCcnt).
// Probe-derived signature: (int4* src_global, __shared__ int4* dst_lds,
//                           imm int offset, imm int cpol)
#if __has_builtin(__builtin_amdgcn_global_load_async_to_lds_b128)
#define HAS_ASYNC_LDS 1
typedef __attribute__((__vector_size__(4 * sizeof(int)))) int v4i_vs;
typedef __attribute__((address_space(3))) v4i_vs lds_v4i;
#else
#define HAS_ASYNC_LDS 0
#endif

#if __has_builtin(__builtin_amdgcn_s_wait_asynccnt)
#define ASYNC_WAIT0() __builtin_amdgcn_s_wait_asynccnt(0)
#else
#define ASYNC_WAIT0() asm volatile("s_wait_asynccnt 0x0" ::: "memory")
#endif

// ---------------------------------------------------------------------------
// Elementwise / reduction kernels
// ---------------------------------------------------------------------------
__global__ void k_symmetrize(const float* __restrict__ W, float* __restrict__ Ws) {
    int idx = blockIdx.x * blockDim.x + threadIdx.x;
    int r = idx >> 10, c = idx & 1023;
    Ws[idx] = 0.5f * (W[idx] + W[c * NN + r]);
}

__global__ void k_init_eye(float* __restrict__ X) {
    int idx = blockIdx.x * blockDim.x + threadIdx.x;
    int r = idx >> 10, c = idx & 1023;
    X[idx] = (r == c) ? 1.0f : 0.0f;
}

// S = X + step*Ws ; partial[block] = sum(S^2) over the block (deterministic)
__global__ void k_step_partial(const float* __restrict__ X, const float* __restrict__ Ws,
                               float* __restrict__ S, float* __restrict__ partial,
                               float step) {
    __shared__ float red[256];
    int idx = blockIdx.x * blockDim.x + threadIdx.x;
    float s = X[idx] + step * Ws[idx];
    S[idx] = s;
    red[threadIdx.x] = s * s;
    __syncthreads();
    for (int off = 128; off > 0; off >>= 1) {
        if ((int)threadIdx.x < off) red[threadIdx.x] += red[threadIdx.x + off];
        __syncthreads();
    }
    if (threadIdx.x == 0) partial[blockIdx.x] = red[0];
}

__global__ void k_reduce_norm(const float* __restrict__ partial, float* __restrict__ normsq,
                              int n) {
    __shared__ float red[256];
    float s = 0.0f;
    for (int i = threadIdx.x; i < n; i += 256) s += partial[i];
    red[threadIdx.x] = s;
    __syncthreads();
    for (int off = 128; off > 0; off >>= 1) {
        if ((int)threadIdx.x < off) red[threadIdx.x] += red[threadIdx.x + off];
        __syncthreads();
    }
    if (threadIdx.x == 0) normsq[0] = red[0];
}

// Z = S / (||S||_F + eps)   (Frobenius >= spectral norm -> Newton-Schulz converges)
__global__ void k_scale(const float* __restrict__ S, const float* __restrict__ normsq,
                        float* __restrict__ Z) {
    int idx = blockIdx.x * blockDim.x + threadIdx.x;
    float c = sqrtf(normsq[0]) + 1e-30f;
    Z[idx] = S[idx] * (1.0f / c);
}

// X = clip(T, 0) off-diagonal, 1 on diagonal   (preserves symmetry)
__global__ void k_finalize(const float* __restrict__ T, float* __restrict__ X) {
    int idx = blockIdx.x * blockDim.x + threadIdx.x;
    int r = idx >> 10, c = idx & 1023;
    X[idx] = (r == c) ? 1.0f : fmaxf(T[idx], 0.0f);
}

__global__ void k_triu(const float* __restrict__ X, float* __restrict__ out) {
    int idx = blockIdx.x * blockDim.x + threadIdx.x;
    int r = idx >> 10, c = idx & 1023;
    out[idx] = (c > r) ? X[idx] : 0.0f;
}

// ---------------------------------------------------------------------------
// FP32 WMMA GEMM for SYMMETRIC B:  Cout = alpha * (A x B) + beta * Cin
//
// B is exactly symmetric in every call of this pipeline, so column n of B
// equals row n: the B tile is staged into LDS as Bt[n][k] via coalesced ROW
// reads -- both A and B fragments then become single aligned ds_load_b64s.
//
// Tile staging uses GLOBAL_LOAD_ASYNC_TO_LDS_B128 with an LDS double buffer
// (ASYNCcnt-tracked; zero VGPR staging pressure -> no spills), falling back
// to direct load+store if the async builtin is unavailable.
// ---------------------------------------------------------------------------
__global__ __launch_bounds__(128)
void k_gemm_wmma_f32(const float* __restrict__ A, const float* __restrict__ B,
                     const float* __restrict__ Cin, float* __restrict__ Cout,
                     float alpha, float beta) {
#if HAS_ASYNC_LDS
    __shared__ float As[2][BM][LDT];   // [buf][m][k]
    __shared__ float Bt[2][BN][LDT];   // [buf][n][k]  (valid because B == B^T)
#else
    __shared__ float As[1][BM][LDT];
    __shared__ float Bt[1][BN][LDT];
#endif

    const int tid  = threadIdx.x;
    const int wave = tid >> 5;
    const int lane = tid & 31;
    const int wm   = (wave >> 1) * 64;   // wave row offset in block tile
    const int wn   = (wave & 1) * 32;    // wave col offset in block tile
    const int lrow = lane & 15;          // M index (A) / N index (B,C)
    const int kgrp = lane >> 4;          // lane-half selects K pair / row half
    const int bm   = blockIdx.y * BM;
    const int bn   = blockIdx.x * BN;

    v8f acc[4][2];
    {
        v8f z = {};
        #pragma unroll
        for (int mt = 0; mt < 4; ++mt)
            #pragma unroll
            for (int nt = 0; nt < 2; ++nt) acc[mt][nt] = z;
    }

    auto computeTile = [&](int buf) {
        #pragma unroll
        for (int ks = 0; ks < BK; ks += 4) {
            const int kb = ks + kgrp * 2;   // this lane-half's K base (K=kb, kb+1)
            v2f a[4], b[2];
            #pragma unroll
            for (int mt = 0; mt < 4; ++mt)
                a[mt] = *(const v2f*)&As[buf][wm + mt * 16 + lrow][kb];
            #pragma unroll
            for (int nt = 0; nt < 2; ++nt)
                b[nt] = *(const v2f*)&Bt[buf][wn + nt * 16 + lrow][kb];
            #pragma unroll
            for (int mt = 0; mt < 4; ++mt)
                #pragma unroll
                for (int nt = 0; nt < 2; ++nt)
                    acc[mt][nt] = __builtin_amdgcn_wmma_f32_16x16x4_f32(
                        false, a[mt], false, b[nt], (short)0, acc[mt][nt],
                        false, false);
        }
    };

    const int NT = NN / BK;   // 32 K-tiles

#if HAS_ASYNC_LDS
    auto issueTile = [&](int t, int buf) {
        const int k0 = t * BK;
        #pragma unroll
        for (int i = 0; i < 8; ++i) {           // A tile: 128 rows x 32 cols
            int q = i * 128 + tid;
            int r = q >> 3, c4 = (q & 7) << 2;
            __builtin_amdgcn_global_load_async_to_lds_b128(
                (v4i_vs*)(A + (size_t)(bm + r) * NN + k0 + c4),
                (lds_v4i*)&As[buf][r][c4], 0, 0);
        }
        #pragma unroll
        for (int i = 0; i < 4; ++i) {           // B tile rows (== columns): 64 x 32
            int q = i * 128 + tid;
            int r = q >> 3, c4 = (q & 7) << 2;
            __builtin_amdgcn_global_load_async_to_lds_b128(
                (v4i_vs*)(B + (size_t)(bn + r) * NN + k0 + c4),
                (lds_v4i*)&Bt[buf][r][c4], 0, 0);
        }
    };

    issueTile(0, 0);
    for (int t = 0; t < NT; t += 2) {
        ASYNC_WAIT0();                 // this wave's async tile writes landed
        __syncthreads();               // all waves' tiles visible
        if (t + 1 < NT) issueTile(t + 1, 1);   // next tile DMAs under compute
        computeTile(0);

        ASYNC_WAIT0();
        __syncthreads();
        if (t + 2 < NT) issueTile(t + 2, 0);
        computeTile(1);
    }
#else
    for (int t = 0; t < NT; ++t) {
        const int k0 = t * BK;
        __syncthreads();
        #pragma unroll
        for (int i = 0; i < 8; ++i) {
            int q = i * 128 + tid;
            int r = q >> 3, c4 = (q & 7) << 2;
            float4 v = *(const float4*)(A + (size_t)(bm + r) * NN + k0 + c4);
            *(float4*)&As[0][r][c4] = v;
        }
        #pragma unroll
        for (int i = 0; i < 4; ++i) {
            int q = i * 128 + tid;
            int r = q >> 3, c4 = (q & 7) << 2;
            float4 v = *(const float4*)(B + (size_t)(bn + r) * NN + k0 + c4);
            *(float4*)&Bt[0][r][c4] = v;
        }
        __syncthreads();
        computeTile(0);
    }
#endif

    // Epilogue: C/D layout -> row_in_tile = v + 8*lane_half, col_in_tile = lane&15
    #pragma unroll
    for (int mt = 0; mt < 4; ++mt) {
        #pragma unroll
        for (int nt = 0; nt < 2; ++nt) {
            #pragma unroll
            for (int v = 0; v < 8; ++v) {
                int row = bm + wm + mt * 16 + v + kgrp * 8;
                int col = bn + wn + nt * 16 + lrow;
                size_t idx = (size_t)row * NN + col;
                float val = alpha * acc[mt][nt][v];
                if (beta != 0.0f) val += beta * Cin[idx];
                Cout[idx] = val;
            }
        }
    }
}

// ---------------------------------------------------------------------------
// Host-side orchestration (all launches on `stream`; graph-capture safe)
// ---------------------------------------------------------------------------
extern "C" void kernel_launch(void* const* d_in, const int* in_sizes, int n_in,
                              void* d_out, int out_size, void* d_ws, size_t ws_size,
                              hipStream_t stream) {
    (void)in_sizes; (void)n_in; (void)out_size; (void)ws_size;

    // inputs: d_in[0] = num_points (scalar int), d_in[1] = edge_weights [1024*1024] f32
    const float* W = (const float*)d_in[1];
    float* out = (float*)d_out;

    const size_t M = (size_t)NN * NN;
    float* Ws      = (float*)d_ws;     // symmetrized weights
    float* X       = Ws + M;           // SDP iterate
    float* S       = X + M;            // pre-projection matrix
    float* Za      = S + M;            // Newton-Schulz ping
    float* Zb      = Za + M;           // Newton-Schulz pong
    float* T       = Zb + M;           // Z^2 / projection temp
    float* partial = T + M;            // 4096 block partial sums
    float* normsq  = partial + 4096;   // scalar ||S||_F^2

    const int  EB = 256;
    const int  EG = (int)(M / EB);     // 4096 blocks
    dim3 gemmGrid(NN / BN, NN / BM);   // 16 x 8 = 128 blocks
    dim3 gemmBlock(128);

    const int OUTER_ITERS = 25;        // reference ITERS
    const int NS_ITERS    = 16;        // Newton-Schulz steps per PSD projection
    const float step      = 1.0f / (float)NN;

    k_symmetrize<<<EG, EB, 0, stream>>>(W, Ws);
    k_init_eye  <<<EG, EB, 0, stream>>>(X);

    for (int it = 0; it < OUTER_ITERS; ++it) {
        // S = X + step * Ws ; normsq = ||S||_F^2 (deterministic 2-stage reduce)
        k_step_partial<<<EG, EB, 0, stream>>>(X, Ws, S, partial, step);
        k_reduce_norm <<<1, 256, 0, stream>>>(partial, normsq, EG);
        // Z0 = S / ||S||_F
        k_scale<<<EG, EB, 0, stream>>>(S, normsq, Za);

        // Newton-Schulz: Z <- 1.5 Z - 0.5 Z Z^2  ==> Z -> sign(S)
        float* zsrc = Za;
        float* zdst = Zb;
        for (int ns = 0; ns < NS_ITERS; ++ns) {
            // T = Z * Z            (B = Z, symmetric)
            k_gemm_wmma_f32<<<gemmGrid, gemmBlock, 0, stream>>>(
                zsrc, zsrc, zsrc, T, 1.0f, 0.0f);
            // Znew = -0.5 * Z * T + 1.5 * Z   (B = T = Z^2, symmetric)
            k_gemm_wmma_f32<<<gemmGrid, gemmBlock, 0, stream>>>(
                zsrc, T, zsrc, zdst, -0.5f, 1.5f);
            float* tmp = zsrc; zsrc = zdst; zdst = tmp;
        }

        // proj_psd(S) = 0.5 * S * sign(S) + 0.5 * S   (B = sign(S), symmetric)
        k_gemm_wmma_f32<<<gemmGrid, gemmBlock, 0, stream>>>(
            S, zsrc, S, T, 0.5f, 0.5f);

        // X = clip(T, 0) with unit diagonal
        k_finalize<<<EG, EB, 0, stream>>>(T, X);
    }

    // out = triu(X, 1)
    k_triu<<<EG, EB, 0, stream>>>(X, out);
}